// GAT_57621281243369
// MI455X (gfx1250) — compile-verified
//
#include <hip/hip_runtime.h>
#include <math.h>

#define N_NODES 50000
#define E_EDGES 800000
#define E_TOT   (E_EDGES + N_NODES)
#define IN_DIM  128
#define HID     128
#define HEADS   8
#define C1      16
#define OUT_DIM 64
#define NEG_SLOPE 0.2f

typedef __attribute__((ext_vector_type(2))) float v2f;
typedef __attribute__((ext_vector_type(8))) float v8f;

static __device__ __forceinline__ unsigned enc_f32(float x) {
    unsigned u = __float_as_uint(x);
    return (u & 0x80000000u) ? ~u : (u | 0x80000000u);
}
static __device__ __forceinline__ float dec_f32(unsigned e) {
    return (e & 0x80000000u) ? __uint_as_float(e ^ 0x80000000u) : __uint_as_float(~e);
}

// One wave computes one 16x16 f32 tile of C = A(MxK) * B(KxLDB), K-step 4 via
// V_WMMA_F32_16X16X4_F32. Block has LDB/16 waves (one per column tile);
// gridDim.x covers row tiles (N_NODES/16 = 3125 exact).
template<int KDIM, int LDB>
__global__ void gemm_wmma(const float* __restrict__ A, const float* __restrict__ B,
                          float* __restrict__ C) {
    const int wave = threadIdx.x >> 5;
    const int lane = threadIdx.x & 31;
    const int row0 = blockIdx.x * 16;
    const int col0 = wave * 16;
    const int m  = lane & 15;
    const int kp = (lane >> 4) << 1;   // half-wave selects K pair {0,1} vs {2,3}

    v8f acc = {};
    const float* arow = A + (size_t)(row0 + m) * KDIM;
    for (int k0 = 0; k0 < KDIM; k0 += 4) {
        v2f a, b;
        a.x = arow[k0 + kp];
        a.y = arow[k0 + kp + 1];
        b.x = B[(size_t)(k0 + kp)     * LDB + col0 + m];
        b.y = B[(size_t)(k0 + kp + 1) * LDB + col0 + m];
        acc = __builtin_amdgcn_wmma_f32_16x16x4_f32(false, a, false, b,
                                                    (short)0, acc, false, false);
    }
    const int rbase = row0 + ((lane >> 4) << 3);
#pragma unroll
    for (int r = 0; r < 8; ++r)
        C[(size_t)(rbase + r) * LDB + col0 + m] = acc[r];
}

// Layer-1 attention scores: one thread per (node, head), C1=16 channels.
__global__ void scores_l1(const float* __restrict__ h, const float* __restrict__ att_s,
                          const float* __restrict__ att_d,
                          float* __restrict__ as, float* __restrict__ ad) {
    int t = blockIdx.x * blockDim.x + threadIdx.x;
    if (t >= N_NODES * HEADS) return;
    int n = t >> 3, hd = t & 7;
    const float* row = h + (size_t)n * HID + hd * C1;
    float s = 0.f, d = 0.f;
#pragma unroll
    for (int c = 0; c < C1; ++c) {
        float v = row[c];
        s += v * att_s[hd * C1 + c];
        d += v * att_d[hd * C1 + c];
    }
    as[t] = s; ad[t] = d;
}

// Layer-2 scores: one wave per node (64 channels, single head), shuffle reduce.
__global__ void scores_l2(const float* __restrict__ h, const float* __restrict__ att_s,
                          const float* __restrict__ att_d,
                          float* __restrict__ as, float* __restrict__ ad) {
    int wave = threadIdx.x >> 5, lane = threadIdx.x & 31;
    int n = blockIdx.x * (blockDim.x >> 5) + wave;
    if (n >= N_NODES) return;
    const float* row = h + (size_t)n * OUT_DIM;
    float s = row[lane] * att_s[lane] + row[lane + 32] * att_s[lane + 32];
    float d = row[lane] * att_d[lane] + row[lane + 32] * att_d[lane + 32];
    for (int off = 16; off; off >>= 1) {
        s += __shfl_xor(s, off, 32);
        d += __shfl_xor(d, off, 32);
    }
    if (lane == 0) { as[n] = s; ad[n] = d; }
}

template<int H>
static __device__ __forceinline__ void edge_sd(const int* __restrict__ ei, int e,
                                               int& s, int& d) {
    if (e < E_EDGES) { s = ei[e]; d = ei[E_EDGES + e]; }
    else             { s = e - E_EDGES; d = s; }
}

template<int H>
__global__ void edge_max(const int* __restrict__ ei, const float* __restrict__ as,
                         const float* __restrict__ ad, unsigned* __restrict__ menc) {
    int t = blockIdx.x * blockDim.x + threadIdx.x;
    if (t >= E_TOT * H) return;
    int e = t / H, hd = t % H, s, d;
    edge_sd<H>(ei, e, s, d);
    float v = as[s * H + hd] + ad[d * H + hd];
    v = v > 0.f ? v : NEG_SLOPE * v;
    atomicMax(&menc[d * H + hd], enc_f32(v));
}

template<int H>
__global__ void edge_expsum(const int* __restrict__ ei, const float* __restrict__ as,
                            const float* __restrict__ ad, const unsigned* __restrict__ menc,
                            float* __restrict__ denom) {
    int t = blockIdx.x * blockDim.x + threadIdx.x;
    if (t >= E_TOT * H) return;
    int e = t / H, hd = t % H, s, d;
    edge_sd<H>(ei, e, s, d);
    float v = as[s * H + hd] + ad[d * H + hd];
    v = v > 0.f ? v : NEG_SLOPE * v;
    float mx = dec_f32(menc[d * H + hd]);
    atomicAdd(&denom[d * H + hd], expf(v - mx));
}

// Scatter alpha-weighted source features into destination accumulators.
// SPLIT threads share one (edge,head); each handles C/SPLIT channels.
template<int H, int C, int SPLIT>
__global__ void edge_scatter(const int* __restrict__ ei, const float* __restrict__ as,
                             const float* __restrict__ ad, const unsigned* __restrict__ menc,
                             const float* __restrict__ denom, const float* __restrict__ hsrc,
                             float* __restrict__ agg) {
    int t = blockIdx.x * blockDim.x + threadIdx.x;
    if (t >= E_TOT * H * SPLIT) return;
    int e   = t / (H * SPLIT);
    int rem = t % (H * SPLIT);
    int hd  = rem / SPLIT;
    int q   = rem % SPLIT;
    int s, d;
    edge_sd<H>(ei, e, s, d);
    float v = as[s * H + hd] + ad[d * H + hd];
    v = v > 0.f ? v : NEG_SLOPE * v;
    float mx = dec_f32(menc[d * H + hd]);
    float alpha = expf(v - mx) / (denom[d * H + hd] + 1e-16f);
    const int CW = C / SPLIT;
    const float* hrow = hsrc + (size_t)s * (H * C) + hd * C + q * CW;
    float* arow       = agg  + (size_t)d * (H * C) + hd * C + q * CW;
#pragma unroll
    for (int c = 0; c < CW; ++c)
        atomicAdd(&arow[c], hrow[c] * alpha);
}

__global__ void elu_bias(float* __restrict__ agg, const float* __restrict__ b) {
    int t = blockIdx.x * blockDim.x + threadIdx.x;
    if (t >= N_NODES * HID) return;
    float v = agg[t] + b[t & (HID - 1)];
    agg[t] = v > 0.f ? v : (expf(v) - 1.0f);
}

// log_softmax over 64 channels: one wave per node, 2 channels per lane.
__global__ void finalize(const float* __restrict__ agg, const float* __restrict__ b,
                         float* __restrict__ out) {
    int wave = threadIdx.x >> 5, lane = threadIdx.x & 31;
    int n = blockIdx.x * (blockDim.x >> 5) + wave;
    if (n >= N_NODES) return;
    float v0 = agg[(size_t)n * OUT_DIM + lane]      + b[lane];
    float v1 = agg[(size_t)n * OUT_DIM + 32 + lane] + b[32 + lane];
    float mx = fmaxf(v0, v1);
    for (int off = 16; off; off >>= 1) mx = fmaxf(mx, __shfl_xor(mx, off, 32));
    float s = expf(v0 - mx) + expf(v1 - mx);
    for (int off = 16; off; off >>= 1) s += __shfl_xor(s, off, 32);
    float lg = mx + logf(s);
    out[(size_t)n * OUT_DIM + lane]      = v0 - lg;
    out[(size_t)n * OUT_DIM + 32 + lane] = v1 - lg;
}

static inline int cdiv(long long a, int b) { return (int)((a + b - 1) / b); }

extern "C" void kernel_launch(void* const* d_in, const int* in_sizes, int n_in,
                              void* d_out, int out_size, void* d_ws, size_t ws_size,
                              hipStream_t stream) {
    (void)in_sizes; (void)n_in; (void)out_size; (void)ws_size;
    const float* x    = (const float*)d_in[0];
    const int*   ei   = (const int*)d_in[1];
    const float* W1   = (const float*)d_in[2];
    const float* aS1  = (const float*)d_in[3];
    const float* aD1  = (const float*)d_in[4];
    const float* b1   = (const float*)d_in[5];
    const float* W2   = (const float*)d_in[6];
    const float* aS2  = (const float*)d_in[7];
    const float* aD2  = (const float*)d_in[8];
    const float* b2   = (const float*)d_in[9];
    float* out = (float*)d_out;

    char* p = (char*)d_ws;
    float*    h1     = (float*)p;    p += (size_t)N_NODES * HID * 4;
    float*    asrc1  = (float*)p;    p += (size_t)N_NODES * HEADS * 4;
    float*    adst1  = (float*)p;    p += (size_t)N_NODES * HEADS * 4;
    unsigned* m1     = (unsigned*)p; p += (size_t)N_NODES * HEADS * 4;   // zeroed region 1 start
    float*    den1   = (float*)p;    p += (size_t)N_NODES * HEADS * 4;
    float*    agg1   = (float*)p;    p += (size_t)N_NODES * HID * 4;     // zeroed region 1 end
    float*    h2     = (float*)p;    p += (size_t)N_NODES * OUT_DIM * 4;
    float*    asrc2  = (float*)p;    p += (size_t)N_NODES * 4;
    float*    adst2  = (float*)p;    p += (size_t)N_NODES * 4;
    unsigned* m2     = (unsigned*)p; p += (size_t)N_NODES * 4;           // zeroed region 2 start
    float*    den2   = (float*)p;    p += (size_t)N_NODES * 4;
    float*    agg2   = (float*)p;    p += (size_t)N_NODES * OUT_DIM * 4; // zeroed region 2 end

    // Zero accumulators / maxima / denominators (must happen every call).
    hipMemsetAsync(m1, 0, (size_t)N_NODES * (HEADS + HEADS + HID) * 4, stream);
    hipMemsetAsync(m2, 0, (size_t)N_NODES * (1 + 1 + OUT_DIM) * 4, stream);

    // ---- Layer 1 ----
    gemm_wmma<IN_DIM, HID><<<N_NODES / 16, (HID / 16) * 32, 0, stream>>>(x, W1, h1);
    scores_l1<<<cdiv((long long)N_NODES * HEADS, 256), 256, 0, stream>>>(h1, aS1, aD1, asrc1, adst1);
    edge_max<HEADS><<<cdiv((long long)E_TOT * HEADS, 256), 256, 0, stream>>>(ei, asrc1, adst1, m1);
    edge_expsum<HEADS><<<cdiv((long long)E_TOT * HEADS, 256), 256, 0, stream>>>(ei, asrc1, adst1, m1, den1);
    edge_scatter<HEADS, C1, 1><<<cdiv((long long)E_TOT * HEADS, 256), 256, 0, stream>>>(
        ei, asrc1, adst1, m1, den1, h1, agg1);
    elu_bias<<<cdiv((long long)N_NODES * HID, 256), 256, 0, stream>>>(agg1, b1);

    // ---- Layer 2 ----
    gemm_wmma<HID, OUT_DIM><<<N_NODES / 16, (OUT_DIM / 16) * 32, 0, stream>>>(agg1, W2, h2);
    scores_l2<<<cdiv(N_NODES, 8), 256, 0, stream>>>(h2, aS2, aD2, asrc2, adst2);
    edge_max<1><<<cdiv(E_TOT, 256), 256, 0, stream>>>(ei, asrc2, adst2, m2);
    edge_expsum<1><<<cdiv(E_TOT, 256), 256, 0, stream>>>(ei, asrc2, adst2, m2, den2);
    edge_scatter<1, OUT_DIM, 4><<<cdiv((long long)E_TOT * 4, 256), 256, 0, stream>>>(
        ei, asrc2, adst2, m2, den2, h2, agg2);
    finalize<<<cdiv(N_NODES, 8), 256, 0, stream>>>(agg2, b2, out);
}